// HashEmbedder_70540542869673
// MI455X (gfx1250) — compile-verified
//
#include <hip/hip_runtime.h>
#include <hip/hip_bf16.h>

#ifndef __has_builtin
#define __has_builtin(x) 0
#endif

#if __has_builtin(__builtin_amdgcn_global_load_async_to_lds_b128)
#define ASYNC_LDS 1
#else
#define ASYNC_LDS 0
#endif

typedef float vf4 __attribute__((ext_vector_type(4)));
typedef int   v4i __attribute__((ext_vector_type(4)));

#define HASH_MASK 0x7FFFFu        // 2^19 - 1
#define PRIME_Y 2654435761u
#define PRIME_Z 805459861u

// int(16 * 1.39**i) for i = 0..15 (matches reference _level_res)
constexpr int kRes[16] = {16, 22, 30, 42, 59, 83, 115, 160,
                          222, 309, 430, 598, 832, 1156, 1608, 2235};
// levels 0..4 are dense ((res+2)^3 < 2^19); 5..15 are hashed

#define L0_ENT 5832               // (16+2)^3 float2 entries  (46,656 B)
#define L1_ENT 13824              // (22+2)^3 float2 entries (110,592 B)
#define LDS_ENT (L0_ENT + L1_ENT) // 19,656 float2 = 157,248 B LDS

struct TabPtrs { const float2* p[16]; };

#if ASYNC_LDS
// Builtin signature (from clang diagnostic): param0 is
// '__attribute__((__vector_size__(4*sizeof(int)))) int __device__ *'
// i.e. int4 in addrspace(1); LDS side is int4 in addrspace(3).
typedef __attribute__((address_space(1))) v4i* g4ptr;
typedef __attribute__((address_space(3))) v4i* l4ptr;
__device__ __forceinline__ void async_cp16(void* lds_dst, const void* g_src) {
  // GLOBAL_LOAD_ASYNC_TO_LDS_B128: 16B global -> LDS, tracked by ASYNCcnt
  __builtin_amdgcn_global_load_async_to_lds_b128(
      (g4ptr)(unsigned long long)g_src,
      (l4ptr)(unsigned)(unsigned long long)lds_dst,
      0, 0);
}
#endif

__device__ __forceinline__ void level_embed(float px, float py, float pz,
                                            int res, bool dense,
                                            const float2* __restrict__ tab,
                                            float& o0, float& o1) {
  const float fr = (float)res;
  float fx = px * fr, fy = py * fr, fz = pz * fr;
  float vx = floorf(fx), vy = floorf(fy), vz = floorf(fz);
  float wx = fx - vx, wy = fy - vy, wz = fz - vz;

  unsigned idx[8];
  if (dense) {
    const int r = res + 2;
    const unsigned ur = (unsigned)r, urr = (unsigned)(r * r);
    unsigned ux = (unsigned)(int)vx, uy = (unsigned)(int)vy, uz = (unsigned)(int)vz;
    unsigned base = uz * urr + uy * ur + ux;
#pragma unroll
    for (int c = 0; c < 8; ++c) {
      // corner c = (i<<2)|(j<<1)|k : i->x(+1), j->y(+r), k->z(+r*r)
      idx[c] = base + (unsigned)((c >> 2) & 1) + (unsigned)((c >> 1) & 1) * ur
             + (unsigned)(c & 1) * urr;
    }
  } else {
    // spatial hash with incremental corner terms (wraparound u32 arithmetic)
    unsigned hx0 = (unsigned)(int)vx,            hx1 = hx0 + 1u;
    unsigned hy0 = (unsigned)(int)vy * PRIME_Y,  hy1 = hy0 + PRIME_Y;
    unsigned hz0 = (unsigned)(int)vz * PRIME_Z,  hz1 = hz0 + PRIME_Z;
#pragma unroll
    for (int c = 0; c < 8; ++c) {
      unsigned h = ((c & 4) ? hx1 : hx0) ^ ((c & 2) ? hy1 : hy0) ^ ((c & 1) ? hz1 : hz0);
      idx[c] = h & HASH_MASK;
    }
  }

  // issue all 8 gathers back-to-back so the wave keeps them in flight
  float2 t[8];
#pragma unroll
  for (int c = 0; c < 8; ++c) t[c] = tab[idx[c]];

  float wxs[2] = {1.f - wx, wx};
  float wys[2] = {1.f - wy, wy};
  float wzs[2] = {1.f - wz, wz};
  float a0 = 0.f, a1 = 0.f;
#pragma unroll
  for (int c = 0; c < 8; ++c) {
    float w = wxs[(c >> 2) & 1] * wys[(c >> 1) & 1] * wzs[c & 1];
    a0 = fmaf(w, t[c].x, a0);
    a1 = fmaf(w, t[c].y, a1);
  }
  o0 = a0; o1 = a1;
}

__global__ __launch_bounds__(256) void HashEmbedder_kernel(
    const float* __restrict__ x, TabPtrs tp, float* __restrict__ out, int n) {
  // Stage the two smallest dense tables (levels 0,1) into LDS: 157,248 B.
  __shared__ float2 lds_tab[LDS_ENT];
  {
    const vf4* g0 = (const vf4*)tp.p[0];
    const vf4* g1 = (const vf4*)tp.p[1];
    vf4* s = (vf4*)lds_tab;
#if ASYNC_LDS
    for (int i = threadIdx.x; i < L0_ENT / 2; i += 256) async_cp16(&s[i], &g0[i]);
    for (int i = threadIdx.x; i < L1_ENT / 2; i += 256)
      async_cp16(&s[L0_ENT / 2 + i], &g1[i]);
#if __has_builtin(__builtin_amdgcn_s_wait_asynccnt)
    __builtin_amdgcn_s_wait_asynccnt(0);
#else
    asm volatile("s_wait_asynccnt 0" ::: "memory");
#endif
#else
    for (int i = threadIdx.x; i < L0_ENT / 2; i += 256) s[i] = g0[i];
    for (int i = threadIdx.x; i < L1_ENT / 2; i += 256) s[L0_ENT / 2 + i] = g1[i];
#endif
  }
  __syncthreads();

  const float2* lds0 = (const float2*)lds_tab;
  const float2* lds1 = (const float2*)lds_tab + L0_ENT;

  int tid = blockIdx.x * blockDim.x + threadIdx.x;
  int stride = gridDim.x * blockDim.x;
  for (int b = tid; b < n; b += stride) {
    // x is streamed once: non-temporal loads keep L2 free for the tables
    float px = __builtin_nontemporal_load(&x[3 * (size_t)b + 0]);
    float py = __builtin_nontemporal_load(&x[3 * (size_t)b + 1]);
    float pz = __builtin_nontemporal_load(&x[3 * (size_t)b + 2]);

    float acc[32];
#pragma unroll
    for (int L = 0; L < 16; ++L) {
      const float2* tab = (L == 0) ? lds0 : (L == 1) ? lds1 : tp.p[L];
      level_embed(px, py, pz, kRes[L], /*dense=*/(L < 5), tab,
                  acc[2 * L], acc[2 * L + 1]);
    }

    // each point owns a contiguous 128B line; stream it out non-temporally
    vf4* o4 = (vf4*)(out + (size_t)b * 32);
#pragma unroll
    for (int q = 0; q < 8; ++q) {
      vf4 v = {acc[4 * q + 0], acc[4 * q + 1], acc[4 * q + 2], acc[4 * q + 3]};
      __builtin_nontemporal_store(v, &o4[q]);
    }
  }
}

extern "C" void kernel_launch(void* const* d_in, const int* in_sizes, int n_in,
                              void* d_out, int out_size, void* d_ws, size_t ws_size,
                              hipStream_t stream) {
  const float* x = (const float*)d_in[0];
  TabPtrs tp;
  for (int i = 0; i < 16; ++i) tp.p[i] = (const float2*)d_in[1 + i];
  int n = in_sizes[0] / 3;  // 2,097,152 points

  // 1024 blocks x 256 threads: ~2048 points/block amortizes the 157KB LDS
  // staging (77 B/pt) against the 128 B/pt of level-0/1 gathers it absorbs,
  // while leaving 2 workgroups per WGP (LDS-limited) = 16 wave32s for latency
  // hiding on the gather stream.
  HashEmbedder_kernel<<<dim3(1024), dim3(256), 0, stream>>>(x, tp, (float*)d_out, n);
}